// SparseLoRAMoE_48954037240489
// MI455X (gfx1250) — compile-verified
//
#include <hip/hip_runtime.h>

typedef float v2f __attribute__((ext_vector_type(2)));
typedef float v8f __attribute__((ext_vector_type(8)));

#define D_IN   2048
#define D_OUT  2048
#define NEXP   16
#define RANK   16
#define KTOP   2
#define ER     256            // NEXP*RANK
#define TOKS   16             // tokens per tile (WMMA M)
#define CHUNK  256            // K-chunk of D_IN staged in LDS
#define XSTR   (CHUNK + 4)    // padded stride -> conflict-free b64 LDS reads
#define HSTR   (ER + 4)

__global__ __launch_bounds__(256)
void moe_lora_wmma_kernel(const float* __restrict__ x,
                          const float* __restrict__ routing,
                          const int*   __restrict__ idxs,
                          const float* __restrict__ w_a,
                          const float* __restrict__ w_b,
                          float* __restrict__ out)
{
    __shared__ float x_lds[TOKS * XSTR];       // 16 x 260 f32
    __shared__ float h_lds[TOKS * HSTR];       // 16 x 260 f32 (gated silu activations)
    __shared__ float g_lds[TOKS * NEXP];       // per-token per-expert gate (incl. alpha=2)

    const int tid  = threadIdx.x;
    const int wave = tid >> 5;
    const int lane = tid & 31;
    const int nIdx = lane & 15;   // N / M index within 16x16 tile
    const int hv   = lane >> 4;   // half-wave select (K-pair / M-offset)
    const int tileBase = blockIdx.x * TOKS;

    // ---- gate table: g[t][e] = 2 * sum_k routing[t,k] * (idx[t,k]==e) ----
    {
        const int t = tid >> 4;          // 0..15  (256 threads == 16*16)
        const int e = tid & 15;
        const int tok = tileBase + t;
        float g = 0.f;
#pragma unroll
        for (int k = 0; k < KTOP; ++k) {
            const int   ie = idxs[tok * KTOP + k];
            const float rw = routing[tok * KTOP + k];
            g += (ie == e) ? rw : 0.f;
        }
        g_lds[t * NEXP + e] = 2.0f * g;
    }

    // ---- phase 1: H[16 x 256] = X[16 x 2048] * W1[2048 x 256] ----
    // wave owns er columns [wave*32, wave*32+32) as two 16-wide N-tiles.
    // 2-way K-split per tile -> 4 independent WMMA accumulation chains.
    v8f acc0a = {}, acc0b = {};
    v8f acc1a = {}, acc1b = {};
    const int er0 = wave * 32;

    for (int c = 0; c < D_IN / CHUNK; ++c) {
        __syncthreads();
        // stage x chunk (16 tokens x 256 dims) into LDS, coalesced float4
        {
            const float* xg = x + (size_t)tileBase * D_IN + c * CHUNK;
#pragma unroll
            for (int i = 0; i < 4; ++i) {
                const int fi  = tid + i * 256;     // float4 id, 0..1023
                const int row = fi >> 6;           // 64 float4 per row
                const int col = (fi & 63) << 2;
                const float4 v = *reinterpret_cast<const float4*>(
                    xg + (size_t)row * D_IN + col);
                *reinterpret_cast<float4*>(&x_lds[row * XSTR + col]) = v;
            }
        }
        __syncthreads();

        const float* waBase = w_a + (size_t)(er0 + nIdx) * D_IN + c * CHUNK;
#pragma unroll 2
        for (int k0 = 0; k0 < CHUNK; k0 += 8) {
            const int kkA = k0 + 2 * hv;
            const int kkB = kkA + 4;
            const v2f a0 = *reinterpret_cast<const v2f*>(&x_lds[nIdx * XSTR + kkA]);
            const v2f a1 = *reinterpret_cast<const v2f*>(&x_lds[nIdx * XSTR + kkB]);
            const v2f b0a = *reinterpret_cast<const v2f*>(waBase + kkA);
            const v2f b1a = *reinterpret_cast<const v2f*>(waBase + (size_t)16 * D_IN + kkA);
            const v2f b0b = *reinterpret_cast<const v2f*>(waBase + kkB);
            const v2f b1b = *reinterpret_cast<const v2f*>(waBase + (size_t)16 * D_IN + kkB);
            acc0a = __builtin_amdgcn_wmma_f32_16x16x4_f32(false, a0, false, b0a,
                                                          (short)0, acc0a, false, false);
            acc1a = __builtin_amdgcn_wmma_f32_16x16x4_f32(false, a0, false, b1a,
                                                          (short)0, acc1a, false, false);
            acc0b = __builtin_amdgcn_wmma_f32_16x16x4_f32(false, a1, false, b0b,
                                                          (short)0, acc0b, false, false);
            acc1b = __builtin_amdgcn_wmma_f32_16x16x4_f32(false, a1, false, b1b,
                                                          (short)0, acc1b, false, false);
        }
    }

    // ---- gate + silu (fast rcp), stash H' in LDS ----
    // C/D layout: element v <-> token (v + 8*hv), column (er0 [+16] + nIdx)
    {
        const v8f h0v = acc0a + acc0b;
        const v8f h1v = acc1a + acc1b;
#pragma unroll
        for (int v = 0; v < 8; ++v) {
            const int t = v + 8 * hv;
            const float h0 = h0v[v];
            const float h1 = h1v[v];
            const float s0 = h0 * __builtin_amdgcn_rcpf(1.0f + __expf(-h0));  // silu
            const float s1 = h1 * __builtin_amdgcn_rcpf(1.0f + __expf(-h1));
            h_lds[t * HSTR + er0 + nIdx]      = g_lds[t * NEXP + wave * 2]     * s0;
            h_lds[t * HSTR + er0 + 16 + nIdx] = g_lds[t * NEXP + wave * 2 + 1] * s1;
        }
    }
    __syncthreads();

    // ---- phase 2: out[16 x 2048] = H'[16 x 256] * W2[256 x 2048] ----
    // W2[er][d] = w_b[e][d][r]; consecutive-r pairs contiguous -> float2 loads.
    // Two d-tiles per iteration (shared A operand) x 2-way K-split -> 4 chains.
    const int dBase = wave * 256;
    for (int nt = 0; nt < 8; ++nt) {
        const int d0 = dBase + nt * 16 + nIdx;
        const int d1 = d0 + 128;
        v8f p0a = {}, p0b = {}, p1a = {}, p1b = {};
#pragma unroll 2
        for (int k0 = 0; k0 < ER; k0 += 8) {
            const int erA = k0 + 2 * hv;           // even -> pair stays in one expert
            const int erB = erA + 4;
            const v2f a0 = *reinterpret_cast<const v2f*>(&h_lds[nIdx * HSTR + erA]);
            const v2f a1 = *reinterpret_cast<const v2f*>(&h_lds[nIdx * HSTR + erB]);
            const int eA = erA >> 4, rA = erA & 15;
            const int eB = erB >> 4, rB = erB & 15;
            const float* wbA = w_b + (size_t)eA * (D_OUT * RANK) + rA;
            const float* wbB = w_b + (size_t)eB * (D_OUT * RANK) + rB;
            const v2f b00 = *reinterpret_cast<const v2f*>(wbA + (size_t)d0 * RANK);
            const v2f b01 = *reinterpret_cast<const v2f*>(wbA + (size_t)d1 * RANK);
            const v2f b10 = *reinterpret_cast<const v2f*>(wbB + (size_t)d0 * RANK);
            const v2f b11 = *reinterpret_cast<const v2f*>(wbB + (size_t)d1 * RANK);
            p0a = __builtin_amdgcn_wmma_f32_16x16x4_f32(false, a0, false, b00,
                                                        (short)0, p0a, false, false);
            p1a = __builtin_amdgcn_wmma_f32_16x16x4_f32(false, a0, false, b01,
                                                        (short)0, p1a, false, false);
            p0b = __builtin_amdgcn_wmma_f32_16x16x4_f32(false, a1, false, b10,
                                                        (short)0, p0b, false, false);
            p1b = __builtin_amdgcn_wmma_f32_16x16x4_f32(false, a1, false, b11,
                                                        (short)0, p1b, false, false);
        }
        const v8f o0 = p0a + p0b;
        const v8f o1 = p1a + p1b;
        float* ob = out + (size_t)tileBase * D_OUT;
#pragma unroll
        for (int v = 0; v < 8; ++v) {
            const size_t row = (size_t)(v + 8 * hv) * D_OUT;
            ob[row + d0] = o0[v];
            ob[row + d1] = o1[v];
        }
    }
}

extern "C" void kernel_launch(void* const* d_in, const int* in_sizes, int n_in,
                              void* d_out, int out_size, void* d_ws, size_t ws_size,
                              hipStream_t stream) {
    const float* x       = (const float*)d_in[0];
    const float* routing = (const float*)d_in[1];
    const int*   idxs    = (const int*)d_in[2];
    const float* w_a     = (const float*)d_in[3];
    const float* w_b     = (const float*)d_in[4];
    float* out = (float*)d_out;

    const int tokens = in_sizes[1] / KTOP;   // routing is [B,S,K] -> B*S tokens
    const int tiles  = tokens / TOKS;        // 4096/16 = 256 blocks

    hipLaunchKernelGGL(moe_lora_wmma_kernel, dim3(tiles), dim3(256), 0, stream,
                       x, routing, idxs, w_a, w_b, out);
}